// HDGE_72481868087309
// MI455X (gfx1250) — compile-verified
//
#include <hip/hip_runtime.h>
#include <hip/hip_bf16.h>
#include <math.h>

// Problem dims (fixed by setup_inputs)
#define N_   128
#define D_   2048
#define C_   256
#define K_   8192
#define INV_T (1.0f / 0.07f)
#define EPS_ 1e-12f

#define LNEG_NSPLIT 2                 // row-range splits per k-slice (occupancy)
#define LNEG_NB_PER (16 / LNEG_NSPLIT)

typedef float v2f __attribute__((ext_vector_type(2)));
typedef float v8f __attribute__((ext_vector_type(8)));

// ---------------------------------------------------------------------------
// FP32 WMMA GEMM:  D[M,N] = A[M,K] @ B[K,N] + bias[N]
// One wave computes one 16x16 tile via V_WMMA_F32_16X16X4_F32.
// A-frag layout (16x4, MxK): lane = (k>=2?16:0)+m ; vgpr0=K{0,2}, vgpr1=K{1,3}
// B-frag layout (4x16, KxN): lanes0-15 hold K0/K1, lanes16-31 hold K2/K3
// C/D layout (16x16): vgpr v, lanes0-15 -> M=v, lanes16-31 -> M=v+8, N=lane&15
// ---------------------------------------------------------------------------
__global__ __launch_bounds__(256) void wmma_gemm_f32(
    const float* __restrict__ A, const float* __restrict__ B,
    const float* __restrict__ bias, float* __restrict__ D,
    int M, int N, int K) {
  const int lane  = threadIdx.x & 31;
  const int wave  = threadIdx.x >> 5;
  const int wid   = blockIdx.x * (blockDim.x >> 5) + wave;
  const int nTile = N >> 4;
  const int mt = wid / nTile;
  const int nt = wid - mt * nTile;
  if (mt * 16 >= M) return;

  const int m     = mt * 16 + (lane & 15);
  const int n     = nt * 16 + (lane & 15);
  const int kband = (lane >> 4) * 2;

  v8f acc;
  const float bv = bias[n];
#pragma unroll
  for (int i = 0; i < 8; ++i) acc[i] = bv;

  const float* __restrict__ aRow = A + (size_t)m * K;

  for (int k = 0; k < K; k += 16) {
#pragma unroll
    for (int kk = 0; kk < 16; kk += 4) {
      const int kb = k + kk + kband;
      v2f a, b;
      a.x = aRow[kb];
      a.y = aRow[kb + 1];
      b.x = B[(size_t)kb * N + n];
      b.y = B[(size_t)(kb + 1) * N + n];
      acc = __builtin_amdgcn_wmma_f32_16x16x4_f32(
          /*neg_a=*/false, a, /*neg_b=*/false, b,
          /*c_mod=*/(short)0, acc, /*reuse_a=*/false, /*reuse_b=*/false);
    }
  }

  const int row0 = mt * 16 + (lane >> 4) * 8;
#pragma unroll
  for (int v = 0; v < 8; ++v) D[(size_t)(row0 + v) * N + n] = acc[v];
}

// ---------------------------------------------------------------------------
// invn[k] = 1 / max(||queue[:,k]||_2, eps)
// ---------------------------------------------------------------------------
__global__ __launch_bounds__(256) void colnorm_kernel(
    const float* __restrict__ q, float* __restrict__ invn) {
  const int k = blockIdx.x * blockDim.x + threadIdx.x;
  float s = 0.f;
#pragma unroll 8
  for (int c = 0; c < C_; ++c) {
    const float v = q[(size_t)c * K_ + k];
    s += v * v;
  }
  invn[k] = 1.0f / fmaxf(sqrtf(s), EPS_);
}

// ---------------------------------------------------------------------------
// l_pos: row-normalize f_logit, weight by dist, logsumexp over C.
// One block per row n (256 threads = C_). Also writes labels[n] = 0.
// ---------------------------------------------------------------------------
__global__ __launch_bounds__(256) void lpos_kernel(
    const float* __restrict__ fl, const float* __restrict__ dist,
    float* __restrict__ out) {
  __shared__ float red[C_];
  __shared__ float bc;
  const int n = blockIdx.x;
  const int c = threadIdx.x;

  const float x = fl[(size_t)n * C_ + c];

  // sum of squares -> row norm
  red[c] = x * x;
  __syncthreads();
#pragma unroll
  for (int s = C_ / 2; s > 0; s >>= 1) {
    if (c < s) red[c] += red[c + s];
    __syncthreads();
  }
  if (c == 0) bc = fmaxf(sqrtf(red[0]), EPS_);
  __syncthreads();

  const float y = dist[(size_t)n * C_ + c] * (x / bc);
  __syncthreads();

  // max reduce
  red[c] = y;
  __syncthreads();
#pragma unroll
  for (int s = C_ / 2; s > 0; s >>= 1) {
    if (c < s) red[c] = fmaxf(red[c], red[c + s]);
    __syncthreads();
  }
  if (c == 0) bc = red[0];
  __syncthreads();
  const float mx = bc;
  __syncthreads();

  // sum exp reduce
  red[c] = __expf(y - mx);
  __syncthreads();
#pragma unroll
  for (int s = C_ / 2; s > 0; s >>= 1) {
    if (c < s) red[c] += red[c + s];
    __syncthreads();
  }
  if (c == 0) {
    out[(size_t)n * (K_ + 1)] = (mx + __logf(red[0])) * INV_T;
    out[(size_t)N_ * (K_ + 1) + n] = 0.0f;  // int32 label 0 == f32 0.0 bits
  }
}

// ---------------------------------------------------------------------------
// l_neg[n,k] = logsumexp_c( dist[n,c] * q[c,k] * invn[k] )
// One block owns a 32-column queue slice: async-copies the 256x32 tile
// (32 KB) into LDS ONCE via GLOBAL_LOAD_ASYNC_TO_LDS_B128, scales it by
// invn in place, then loops over 8-row blocks of dist (8 KB tiles),
// doing two LDS-only passes (max, then sum-exp) per row block.
// ---------------------------------------------------------------------------
__global__ __launch_bounds__(256) void lneg_kernel(
    const float* __restrict__ dist, const float* __restrict__ q,
    const float* __restrict__ invn, float* __restrict__ out) {
  __shared__ float qs[C_][32];  // 32 KB queue tile (pre-scaled by invn)
  __shared__ float ds[8][C_];   //  8 KB dist tile

  const int t     = threadIdx.x;
  const int kb    = blockIdx.x >> 1;            // 256 k-slices
  const int split = blockIdx.x & (LNEG_NSPLIT - 1);
  const int k0    = kb * 32;
  const int kk    = t & 31;
  const int nl    = t >> 5;

  // --- async global -> LDS copy of the queue tile (16 B per lane per op) ---
  // 2048 float4 chunks, 256 threads -> 8 ops/thread. Generic pointer low 32
  // bits == LDS byte offset (flat aperture rule), which is what VDST wants.
  for (int i = t; i < (C_ * 32) / 4; i += 256) {
    const int c   = i >> 3;          // 8 float4 per 32-float row
    const int kk4 = (i & 7) * 4;
    const unsigned lds_off = (unsigned)(size_t)&qs[c][kk4];
    const float* gp = q + (size_t)c * K_ + k0 + kk4;
    asm volatile("global_load_async_to_lds_b128 %0, %1, off"
                 :: "v"(lds_off), "v"(gp) : "memory");
  }
  asm volatile("s_wait_asynccnt 0x0" ::: "memory");
  __syncthreads();

  // in-place scale by column inverse norm (each thread owns column kk)
  const float inv_k = invn[k0 + kk];
  for (int i = t; i < C_ * 32; i += 256) qs[i >> 5][kk] *= inv_k;

  // --- loop over 8-row blocks of dist ---
  for (int nb = split * LNEG_NB_PER; nb < (split + 1) * LNEG_NB_PER; ++nb) {
    const int n0 = nb * 8;
    __syncthreads();  // previous iteration's compute done before overwrite
    for (int i = t; i < 8 * C_; i += 256) {
      const int r = i >> 8, c = i & (C_ - 1);
      ds[r][c] = dist[(size_t)(n0 + r) * C_ + c];
    }
    __syncthreads();

    float mx = -3.4e38f;
#pragma unroll 8
    for (int c = 0; c < C_; ++c) mx = fmaxf(mx, ds[nl][c] * qs[c][kk]);

    float s = 0.f;
#pragma unroll 8
    for (int c = 0; c < C_; ++c) s += __expf(ds[nl][c] * qs[c][kk] - mx);

    out[(size_t)(n0 + nl) * (K_ + 1) + 1 + k0 + kk] =
        (mx + __logf(s)) * INV_T;
  }
}

// ---------------------------------------------------------------------------
extern "C" void kernel_launch(void* const* d_in, const int* in_sizes, int n_in,
                              void* d_out, int out_size, void* d_ws, size_t ws_size,
                              hipStream_t stream) {
  (void)in_sizes; (void)n_in; (void)out_size; (void)ws_size;
  const float* img  = (const float*)d_in[0];  // [128,2048]
  const float* W_f  = (const float*)d_in[1];  // [2048,256]
  const float* b_f  = (const float*)d_in[2];  // [256]
  const float* W_c  = (const float*)d_in[3];  // [256,256]
  const float* b_c  = (const float*)d_in[4];  // [256]
  const float* dist = (const float*)d_in[5];  // [128,256]
  const float* q    = (const float*)d_in[6];  // [256,8192]

  float* out     = (float*)d_out;
  float* penult  = (float*)d_ws;                      // 128*256 floats
  float* invn    = penult + N_ * C_;                  // 8192 floats
  float* f_logit = out + (size_t)N_ * (K_ + 1) + N_;  // third output region

  // penult = img @ W_f + b_f : 128 tiles -> 128 waves -> 16 blocks of 8 waves
  wmma_gemm_f32<<<16, 256, 0, stream>>>(img, W_f, b_f, penult, N_, C_, D_);
  // f_logit = penult @ W_c + b_c (written straight into d_out region)
  wmma_gemm_f32<<<16, 256, 0, stream>>>(penult, W_c, b_c, f_logit, N_, C_, C_);
  // queue column inverse norms
  colnorm_kernel<<<K_ / 256, 256, 0, stream>>>(q, invn);
  // positive logit + labels
  lpos_kernel<<<N_, 256, 0, stream>>>(f_logit, dist, out);
  // negative logits (bulk of the work): 256 k-slices x 2 row splits
  lneg_kernel<<<256 * LNEG_NSPLIT, 256, 0, stream>>>(dist, q, invn, out);
}